// TransRModel_49512382988421
// MI455X (gfx1250) — compile-verified
//
#include <hip/hip_runtime.h>
#include <hip/hip_bf16.h>
#include <math.h>

// TransR margin loss on MI455X (gfx1250, wave32).
//
// Key algebra: head_p - tail_p = M @ (head - tail), so ONE fp32 matvec per
// triple: score = sum_j | (M @ (h - t))[j] + rel[j] |.
// One wave per (side, triple). The matvec runs on the fp32 matrix pipe via
// V_WMMA_F32_16X16X4_F32 with B column 0 = (head - tail) chunk, all other
// columns zero. Memory bound: 16384 triples x 40KB of M = 655 MB streamed
// from the 192MB L2 (40MB footprint); compute is ~0.33 GFLOP -> keep fp32.
#define DIM    100
#define NBATCH 8192

typedef __attribute__((ext_vector_type(2))) float v2f;
typedef __attribute__((ext_vector_type(8))) float v8f;

__global__ __launch_bounds__(128) void transr_score_kernel(
    const int* __restrict__ pos_tri, const int* __restrict__ neg_tri,
    const float* __restrict__ ent,   const float* __restrict__ rel,
    const float* __restrict__ transfer, float* __restrict__ scores)
{
    const int widx = (blockIdx.x * blockDim.x + threadIdx.x) >> 5; // global wave
    const int lane = threadIdx.x & 31;
    const int side = widx >> 13;            // 0 = pos, 1 = neg
    const int b    = widx & (NBATCH - 1);
    const int* __restrict__ tri = side ? neg_tri : pos_tri;

    const int h = tri[b * 3 + 0];
    const int r = tri[b * 3 + 1];
    const int t = tri[b * 3 + 2];

    const float* __restrict__ M  = transfer + (size_t)r * (DIM * DIM);
    const float* __restrict__ hv = ent + (size_t)h * DIM;
    const float* __restrict__ tv = ent + (size_t)t * DIM;
    const float* __restrict__ rv = rel + (size_t)r * DIM;

    // ISA 7.12.2 fp32 operand lanes:
    //   A 16x4 : lanes 0-15 row m=lane, K={0,1}; lanes 16-31 row m=lane-16, K={2,3}
    //   B 4x16 : column n = lane&15 (only n==0 live here)
    //   D 16x16: VGPR i -> row i (lanes 0-15) / row i+8 (lanes 16-31)
    const int   n       = lane & 15;
    const int   khalf   = (lane >> 4) << 1;     // 0 or 2
    const int   rowoff  = (lane >> 4) << 3;     // 0 or 8 (D row offset)
    const float colmask = (n == 0) ? 1.0f : 0.0f;

    // Hoist B: per-lane (head - tail) chunks, masked to column 0, kept in VGPRs
    // and reused across all 7 row blocks (branchless; every lane loads a valid
    // address, the multiply zeroes dead columns).
    v2f dreg[25];
#pragma unroll
    for (int kc = 0; kc < 25; ++kc) {
        const int    k0 = kc * 4 + khalf;
        const float2 hh = *(const float2*)(hv + k0);
        const float2 tt = *(const float2*)(tv + k0);
        v2f d;
        d.x = (hh.x - tt.x) * colmask;
        d.y = (hh.y - tt.y) * colmask;
        dreg[kc] = d;
    }

    float score = 0.0f;

    // Row blocks 0..5: all 16 rows valid, no masking anywhere.
    for (int rb = 0; rb < 6; ++rb) {
        const float* __restrict__ Arow = M + (size_t)(rb * 16 + n) * DIM;
        __builtin_prefetch(Arow + 16 * DIM, 0, 3);   // next row block of M

        v8f acc = {0.f, 0.f, 0.f, 0.f, 0.f, 0.f, 0.f, 0.f};
#pragma unroll
        for (int kc = 0; kc < 25; ++kc) {
            const float2 av = *(const float2*)(Arow + kc * 4 + khalf);
            v2f a; a.x = av.x; a.y = av.y;
            acc = __builtin_amdgcn_wmma_f32_16x16x4_f32(
                false, a, false, dreg[kc], (short)0, acc, false, false);
        }
#pragma unroll
        for (int i = 0; i < 8; ++i) {
            const int row = rb * 16 + rowoff + i;
            score += colmask * fabsf(acc[i] + rv[row]);
        }
    }

    // Row block 6: rows 96..111, only 96..99 valid -> clamp + float mask
    // (branchless, no EXEC manipulation).
    {
        const int   arow  = 96 + n;
        const float amask = (arow < DIM) ? 1.0f : 0.0f;
        const float* __restrict__ Arow =
            M + (size_t)((arow < DIM) ? arow : (DIM - 1)) * DIM;

        v8f acc = {0.f, 0.f, 0.f, 0.f, 0.f, 0.f, 0.f, 0.f};
#pragma unroll
        for (int kc = 0; kc < 25; ++kc) {
            const float2 av = *(const float2*)(Arow + kc * 4 + khalf);
            v2f a; a.x = av.x * amask; a.y = av.y * amask;
            acc = __builtin_amdgcn_wmma_f32_16x16x4_f32(
                false, a, false, dreg[kc], (short)0, acc, false, false);
        }
#pragma unroll
        for (int i = 0; i < 8; ++i) {
            const int   row   = 96 + rowoff + i;
            const float vmask = (row < DIM) ? colmask : 0.0f;
            const int   rowc  = (row < DIM) ? row : (DIM - 1);
            score += vmask * fabsf(acc[i] + rv[rowc]);
        }
    }

    // Only lanes 0 and 16 carry nonzero partials (colmask): one shuffle folds
    // lane 16 into lane 0.
    score += __shfl_down(score, 16, 32);
    if (lane == 0) scores[widx] = score;
}

__global__ __launch_bounds__(1024) void transr_margin_reduce(
    const float* __restrict__ scores, float* __restrict__ out)
{
    __shared__ float sdata[32];
    const int lane = threadIdx.x & 31;
    const int wid  = threadIdx.x >> 5;

    float local = 0.0f;
    for (int b = threadIdx.x; b < NBATCH; b += 1024)
        local += fmaxf(0.0f, scores[b] - scores[NBATCH + b] + 1.0f);

    for (int off = 16; off > 0; off >>= 1)
        local += __shfl_down(local, off, 32);
    if (lane == 0) sdata[wid] = local;
    __syncthreads();

    if (wid == 0) {
        float v = sdata[lane];              // exactly 32 waves in block
        for (int off = 16; off > 0; off >>= 1)
            v += __shfl_down(v, off, 32);
        if (lane == 0) out[0] = v * (1.0f / (float)NBATCH);
    }
}

extern "C" void kernel_launch(void* const* d_in, const int* in_sizes, int n_in,
                              void* d_out, int out_size, void* d_ws, size_t ws_size,
                              hipStream_t stream)
{
    (void)in_sizes; (void)n_in; (void)out_size; (void)ws_size;

    const int*   pos      = (const int*)d_in[0];   // pos_triples (8192 x 3)
    const int*   neg      = (const int*)d_in[1];   // neg_triples (8192 x 3)
    const float* ent      = (const float*)d_in[2]; // ent_emb (500000 x 100)
    const float* rel      = (const float*)d_in[3]; // rel_emb (1000 x 100)
    const float* transfer = (const float*)d_in[4]; // transfer (1000 x 10000)

    float* scores = (float*)d_ws;                  // 2 * 8192 floats of scratch

    // 16384 waves (pos+neg), 4 waves / 128-thread block -> 4096 blocks.
    transr_score_kernel<<<4096, 128, 0, stream>>>(pos, neg, ent, rel,
                                                  transfer, scores);
    transr_margin_reduce<<<1, 1024, 0, stream>>>(scores, (float*)d_out);
}